// DigitCapsule_24936580121369
// MI455X (gfx1250) — compile-verified
//
#include <hip/hip_runtime.h>
#include <math.h>

// ---------------------------------------------------------------------------
// CapsNet dynamic routing, fully fused for MI455X (gfx1250).
//   votes[b,i,j,d] = W[i,j,d,:] . x[b,i,:]   (never materialized: 189MB saved)
//   3 routing iterations, each: c=softmax_j(b), s=sum_i c*votes, v=squash(s),
//   b += votes.v   where b is recomputed on the fly as votes.(v0[+v1]).
// Core compute: per (i, 16-batch tile) a 16x8 @ 8x160 GEMM done with
// V_WMMA_F32_16X16X4_F32, N ordered as j*16+d so each 16-col WMMA tile is one
// digit capsule j. D-layout: lane%16 = d, vgprRow + 8*(lane/16) = local batch.
// ---------------------------------------------------------------------------

typedef float v2f __attribute__((ext_vector_type(2)));
typedef float v8f __attribute__((ext_vector_type(8)));

#define B_TOT   256
#define I_TOT   1152
#define P_DIM   8
#define J_NUM   10
#define D_DIM   16
#define JD      160          // J_NUM * D_DIM
#define BTILE   16           // batch rows per WMMA tile
#define NWAVES  4
#define BLOCK   (NWAVES * 32)
#define ISPLIT  36           // i-chunks in grid.y
#define ICHUNK  (I_TOT / ISPLIT)   // 32 i's per workgroup

// ---------------------------------------------------------------------------
// Accumulation kernel: one workgroup = (16-batch tile) x (32-i chunk).
// Each wave recomputes votes tiles with WMMA, derives routing coefficients,
// and atomically accumulates s[b,j,d].
// r = routing iteration (0,1,2). vh0/vh1 = squash outputs of prior iterations.
// ---------------------------------------------------------------------------
__global__ __launch_bounds__(BLOCK)
void caps_route_accum(const float* __restrict__ x,   // [B,I,P]
                      const float* __restrict__ W,   // [I,J,D,P]
                      const float* __restrict__ vh0, // [B,J,D]
                      const float* __restrict__ vh1, // [B,J,D]
                      float* __restrict__ s,         // [B,J,D] (pre-zeroed)
                      int r)
{
    __shared__ float lds_v[BTILE * JD];   // v0 (+ v1) summed, for this b-tile

    const int tid  = threadIdx.x;
    const int lane = tid & 31;
    const int wave = tid >> 5;
    const int n    = lane & 15;   // A-row (local batch) AND B-col (d index)
    const int h    = lane >> 4;   // lane half selects K sub-rows
    const int b0   = blockIdx.x * BTILE;
    const int ibase = blockIdx.y * ICHUNK;

    // Stage summed v-history: logits b = votes . (v0 [+ v1])
    if (r > 0) {
        for (int idx = tid; idx < BTILE * JD; idx += BLOCK) {
            const int gb = (b0 + idx / JD) * JD + (idx % JD);
            float t = vh0[gb];
            if (r == 2) t += vh1[gb];
            lds_v[idx] = t;
        }
    }
    __syncthreads();

    v8f s_acc[J_NUM];
#pragma unroll
    for (int j = 0; j < J_NUM; ++j) s_acc[j] = (v8f){};

    for (int ii = wave; ii < ICHUNK; ii += NWAVES) {
        const int i = ibase + ii;

        // ---- A operand: x tile, 16x8. VGPR0 holds K={0,2(+kbase)}, VGPR1 K={1,3}.
        const float* xp = x + ((size_t)(b0 + n) * I_TOT + i) * P_DIM;
        const v2f a0 = *(const v2f*)(xp + 2 * h);       // K = 0..3
        const v2f a1 = *(const v2f*)(xp + 4 + 2 * h);   // K = 4..7

        // ---- votes tiles: one 16x16 WMMA tile per digit capsule j (K=8 in 2 steps)
        v8f votes[J_NUM];
        const float* wp = W + (size_t)i * J_NUM * D_DIM * P_DIM;
#pragma unroll
        for (int j = 0; j < J_NUM; ++j) {
            const float* wjp = wp + j * D_DIM * P_DIM + n * P_DIM;
            const v2f bm0 = *(const v2f*)(wjp + 2 * h);
            const v2f bm1 = *(const v2f*)(wjp + 4 + 2 * h);
            v8f acc = (v8f){};
            acc = __builtin_amdgcn_wmma_f32_16x16x4_f32(
                      false, a0, false, bm0, (short)0, acc, false, false);
            acc = __builtin_amdgcn_wmma_f32_16x16x4_f32(
                      false, a1, false, bm1, (short)0, acc, false, false);
            votes[j] = acc;
        }

        if (r == 0) {
            // b == 0 -> softmax is uniform 1/J
#pragma unroll
            for (int j = 0; j < J_NUM; ++j) s_acc[j] += 0.1f * votes[j];
        } else {
            // ---- logits: blog[v][j] = sum_d votes[b,i,j,d] * vsum[b,j,d]
            // d lives on lanes within a 16-lane half -> xor-shuffle reduce.
            float blog[8][J_NUM];
#pragma unroll
            for (int v = 0; v < 8; ++v) {
                const int brow = v + 8 * h;
#pragma unroll
                for (int j = 0; j < J_NUM; ++j) {
                    float t = votes[j][v] * lds_v[brow * JD + j * D_DIM + n];
                    t += __shfl_xor(t, 1, 32);
                    t += __shfl_xor(t, 2, 32);
                    t += __shfl_xor(t, 4, 32);
                    t += __shfl_xor(t, 8, 32);
                    blog[v][j] = t;   // replicated across the 16-lane half
                }
            }
            // ---- softmax over J per (batch-row v), then weighted s accumulation
#pragma unroll
            for (int v = 0; v < 8; ++v) {
                float m = blog[v][0];
#pragma unroll
                for (int j = 1; j < J_NUM; ++j) m = fmaxf(m, blog[v][j]);
                float e[J_NUM];
                float sum = 0.f;
#pragma unroll
                for (int j = 0; j < J_NUM; ++j) {
                    e[j] = __expf(blog[v][j] - m);
                    sum += e[j];
                }
                const float inv = __frcp_rn(sum);
#pragma unroll
                for (int j = 0; j < J_NUM; ++j) {
                    s_acc[j][v] += (e[j] * inv) * votes[j][v];
                }
            }
        }
    }

    // ---- flush per-wave partial s into global accumulator
#pragma unroll
    for (int j = 0; j < J_NUM; ++j) {
#pragma unroll
        for (int v = 0; v < 8; ++v) {
            atomicAdd(&s[(size_t)(b0 + v + 8 * h) * JD + j * D_DIM + n],
                      s_acc[j][v]);
        }
    }
}

// ---------------------------------------------------------------------------
__global__ void caps_zero(float* __restrict__ p, int nelem)
{
    const int idx = blockIdx.x * blockDim.x + threadIdx.x;
    if (idx < nelem) p[idx] = 0.f;
}

// squash: v = s * ||s||^2 / (1+||s||^2) / sqrt(||s||^2 + eps), over D=16
// One thread per element; d == idx%16 aligns with 16-lane shuffle groups.
__global__ void caps_squash(const float* __restrict__ s,
                            float* __restrict__ vout, int nelem)
{
    const int idx = blockIdx.x * blockDim.x + threadIdx.x;
    if (idx >= nelem) return;
    const float val = s[idx];
    float s2 = val * val;
    s2 += __shfl_xor(s2, 1, 32);
    s2 += __shfl_xor(s2, 2, 32);
    s2 += __shfl_xor(s2, 4, 32);
    s2 += __shfl_xor(s2, 8, 32);
    const float scale = s2 / (1.f + s2) / sqrtf(s2 + 1e-7f);
    vout[idx] = val * scale;
}

// ---------------------------------------------------------------------------
extern "C" void kernel_launch(void* const* d_in, const int* in_sizes, int n_in,
                              void* d_out, int out_size, void* d_ws, size_t ws_size,
                              hipStream_t stream)
{
    (void)in_sizes; (void)n_in; (void)ws_size;
    const float* x = (const float*)d_in[0];   // [256,1152,8]
    const float* W = (const float*)d_in[1];   // [1152,10,16,8]
    float* out = (float*)d_out;               // [256,10,16]

    const int NSJD = B_TOT * JD;              // 40960 elements
    float* s  = (float*)d_ws;                 // [B,J,D] accumulator
    float* v0 = s  + NSJD;                    // v after iter 0
    float* v1 = v0 + NSJD;                    // v after iter 1

    const dim3 agrid(B_TOT / BTILE, ISPLIT);  // 16 x 36 workgroups
    const int  ZB = (NSJD + 255) / 256;

    for (int r = 0; r < 3; ++r) {
        caps_zero<<<ZB, 256, 0, stream>>>(s, NSJD);
        caps_route_accum<<<agrid, BLOCK, 0, stream>>>(x, W, v0, v1, s, r);
        float* vdst = (r == 2) ? out : (r == 0 ? v0 : v1);
        caps_squash<<<ZB, 256, 0, stream>>>(s, vdst, NSJD);
    }
    (void)out_size;
}